// GNN_node_Virtualnode_60490319397094
// MI455X (gfx1250) — compile-verified
//
#include <hip/hip_runtime.h>
#include <hip/hip_bf16.h>

typedef __attribute__((ext_vector_type(2))) float v2f;
typedef __attribute__((ext_vector_type(8))) float v8f;

#define Dn   128
#define D2n  256
#define Nn   20000
#define En   640000
#define Gn   128
#define Ln   5
#define EDn  7
#define BN_EPS 1e-5f

// ---------------------------------------------------------------------------
// WMMA fp32 GEMM: C[M,NC] = op(A)[M,K] @ W[K,NC] + bias[NC]
// op(A) = identity, or relu(ac[k]*A + cc[k]) (fused BN+ReLU of previous stage)
// One wave per 16x16 tile, K stepped by 4 via v_wmma_f32_16x16x4_f32.
// ---------------------------------------------------------------------------
template <bool TRANS>
__global__ void k_gemm_wmma(const float* __restrict__ A, const float* __restrict__ W,
                            const float* __restrict__ bias, float* __restrict__ C,
                            int M, int K, int NC,
                            const float* __restrict__ ac, const float* __restrict__ cc)
{
    int wave = threadIdx.x >> 5;
    int lane = threadIdx.x & 31;
    int tile = blockIdx.x * (blockDim.x >> 5) + wave;
    int tn = NC >> 4;
    int ntiles = (M >> 4) * tn;
    if (tile >= ntiles) return;                    // wave-uniform: EXEC stays all-1s
    int tm = tile / tn;
    int tc = tile - tm * tn;
    int row0 = tm << 4, col0 = tc << 4;
    int half = lane >> 4, l16 = lane & 15;

    v8f acc;
    float bv = bias[col0 + l16];                   // bias depends on column only
#pragma unroll
    for (int i = 0; i < 8; ++i) acc[i] = bv;

    // A 16x4 fragment: lane half 0 -> K={k,k+1}, half 1 -> K={k+2,k+3}
    const float* ap = A + (size_t)(row0 + l16) * K + 2 * half;
    const float* wp = W + (size_t)(2 * half) * NC + col0 + l16;

    for (int k = 0; k < K; k += 4) {
        v2f a, b;
        a.x = ap[0];
        a.y = ap[1];
        if (TRANS) {
            int ka = k + 2 * half;
            a.x = fmaxf(ac[ka]     * a.x + cc[ka],     0.f);
            a.y = fmaxf(ac[ka + 1] * a.y + cc[ka + 1], 0.f);
        }
        b.x = wp[0];
        b.y = wp[NC];
        acc = __builtin_amdgcn_wmma_f32_16x16x4_f32(false, a, false, b,
                                                    (short)0, acc, false, false);
        ap += 4;
        wp += 4 * NC;
    }

    // D layout: VGPR i -> row (row0 + i + 8*half), col (col0 + l16)
    float* cp = C + (size_t)(row0 + half * 8) * NC + col0 + l16;
#pragma unroll
    for (int i = 0; i < 8; ++i) cp[(size_t)i * NC] = acc[i];
}

// vn init: vn[g,c] = vn_emb[c]
__global__ void k_vn_init(const float* __restrict__ vn_emb, float* __restrict__ vn, int total)
{
    int i = blockIdx.x * blockDim.x + threadIdx.x;
    if (i < total) vn[i] = vn_emb[i & (Dn - 1)];
}

// h_in = h + vn[batch];  Z = (1+eps_l)*h_in  (Z is the scatter-add target -> becomes z)
__global__ void k_hin(const float* __restrict__ h, const float* __restrict__ vn,
                      const int* __restrict__ batch, const float* __restrict__ eps_l,
                      float* __restrict__ h_in, float* __restrict__ Zb, int total)
{
    int i = blockIdx.x * blockDim.x + threadIdx.x;
    if (i >= total) return;
    int nd = i >> 7, c = i & (Dn - 1);
    float v = h[i] + vn[batch[nd] * Dn + c];
    h_in[i] = v;
    Zb[i] = (1.0f + eps_l[0]) * v;
}

// One wave per edge: eemb = edge_attr@Wl + bl (Wl in LDS), msg = relu(h_in[src]+eemb),
// atomic f32 scatter-add into Z[dst].
__global__ void k_edge(const float* __restrict__ edge_attr, const int* __restrict__ eidx,
                       const float* __restrict__ Wl, const float* __restrict__ bl,
                       const float* __restrict__ h_in, float* __restrict__ Z, int E_)
{
    __shared__ float sW[EDn * Dn];
    __shared__ float sB[Dn];
    for (int i = threadIdx.x; i < EDn * Dn; i += blockDim.x) sW[i] = Wl[i];
    if (threadIdx.x < Dn) sB[threadIdx.x] = bl[threadIdx.x];
    __syncthreads();

    int gid = blockIdx.x * blockDim.x + threadIdx.x;
    int e = gid >> 5;
    int lane = threadIdx.x & 31;
    if (e >= E_) return;

    int s = eidx[e];
    int d = eidx[E_ + e];
    float ea[EDn];
#pragma unroll
    for (int j = 0; j < EDn; ++j) ea[j] = edge_attr[(size_t)e * EDn + j];

    const float4 hv = *(const float4*)(h_in + (size_t)s * Dn + lane * 4);
    float hq[4] = {hv.x, hv.y, hv.z, hv.w};
#pragma unroll
    for (int q = 0; q < 4; ++q) {
        int c = lane * 4 + q;
        float em = sB[c];
#pragma unroll
        for (int j = 0; j < EDn; ++j) em += ea[j] * sW[j * Dn + c];
        float v = fmaxf(hq[q] + em, 0.f);
        atomicAdd(&Z[(size_t)d * Dn + c], v);
    }
}

// column-wise sum / sumsq (blockDim == C, coalesced rows, grid-strided)
__global__ void k_colstats(const float* __restrict__ X, int Mrows, int C,
                           float* __restrict__ sums, float* __restrict__ sqs)
{
    int c = threadIdx.x;
    float s = 0.f, q = 0.f;
    for (int r = blockIdx.x; r < Mrows; r += gridDim.x) {
        float v = X[(size_t)r * C + c];
        s += v;
        q += v * v;
    }
    atomicAdd(&sums[c], s);
    atomicAdd(&sqs[c], q);
}

// per-channel BN affine coefficients: y = a*x + c
__global__ void k_bncoef(const float* __restrict__ sums, const float* __restrict__ sqs,
                         const float* __restrict__ g, const float* __restrict__ b,
                         int C, float invM, float* __restrict__ ac, float* __restrict__ cc)
{
    int c = blockIdx.x * blockDim.x + threadIdx.x;
    if (c >= C) return;
    float m = sums[c] * invM;
    float v = sqs[c] * invM - m * m;
    float inv = rsqrtf(v + BN_EPS);
    float a = g[c] * inv;
    ac[c] = a;
    cc[c] = b[c] - m * a;
}

// Y = maybe_relu(a[col]*X + c[col])
__global__ void k_affine(const float* __restrict__ X, const float* __restrict__ a,
                         const float* __restrict__ c, float* __restrict__ Y,
                         int total, int C, int dorelu)
{
    int i = blockIdx.x * blockDim.x + threadIdx.x;
    if (i >= total) return;
    int col = i & (C - 1);
    float v = a[col] * X[i] + c[col];
    if (dorelu) v = fmaxf(v, 0.f);
    Y[i] = v;
}

// pooled = vn
__global__ void k_pool_init(const float* __restrict__ vn, float* __restrict__ pooled, int total)
{
    int i = blockIdx.x * blockDim.x + threadIdx.x;
    if (i < total) pooled[i] = vn[i];
}

// pooled[batch[n]] += h_in[n]
__global__ void k_pool_atomic(const float* __restrict__ h_in, const int* __restrict__ batch,
                              float* __restrict__ pooled, int total)
{
    int i = blockIdx.x * blockDim.x + threadIdx.x;
    if (i >= total) return;
    int nd = i >> 7, c = i & (Dn - 1);
    atomicAdd(&pooled[batch[nd] * Dn + c], h_in[i]);
}

// virtual-node MLP stage: out[G,C] = relu(BN_over_G(X@W + b))
// one block per output column, thread r = graph row (G == 128 == blockDim)
__global__ void k_vn_layer(const float* __restrict__ X, int K,
                           const float* __restrict__ W, int C,
                           const float* __restrict__ bias,
                           const float* __restrict__ gam, const float* __restrict__ bet,
                           float* __restrict__ out)
{
    int c = blockIdx.x;
    int r = threadIdx.x;   // 0..127
    float acc = bias[c];
    for (int k = 0; k < K; ++k) acc += X[(size_t)r * K + k] * W[(size_t)k * C + c];

    __shared__ float ssum[Gn];
    __shared__ float ssq[Gn];
    ssum[r] = acc;
    ssq[r]  = acc * acc;
    __syncthreads();
    for (int s = Gn / 2; s > 0; s >>= 1) {
        if (r < s) { ssum[r] += ssum[r + s]; ssq[r] += ssq[r + s]; }
        __syncthreads();
    }
    float m = ssum[0] * (1.0f / Gn);
    float v = ssq[0] * (1.0f / Gn) - m * m;
    float inv = rsqrtf(v + BN_EPS);
    float y = gam[c] * (acc - m) * inv + bet[c];
    out[(size_t)r * C + c] = fmaxf(y, 0.f);
}

extern "C" void kernel_launch(void* const* d_in, const int* in_sizes, int n_in,
                              void* d_out, int out_size, void* d_ws, size_t ws_size,
                              hipStream_t stream)
{
    const float* x         = (const float*)d_in[0];
    const float* edge_attr = (const float*)d_in[1];
    const int*   eidx      = (const int*)d_in[2];
    const int*   batch     = (const int*)d_in[3];
    const float* enc_W     = (const float*)d_in[4];
    const float* enc_b     = (const float*)d_in[5];
    const float* vn_emb    = (const float*)d_in[6];
    const float* eps       = (const float*)d_in[7];
    const float* edge_W    = (const float*)d_in[8];
    const float* edge_b    = (const float*)d_in[9];
    const float* mlp_W1    = (const float*)d_in[10];
    const float* mlp_b1    = (const float*)d_in[11];
    const float* mlp_bn1_g = (const float*)d_in[12];
    const float* mlp_bn1_b = (const float*)d_in[13];
    const float* mlp_W2    = (const float*)d_in[14];
    const float* mlp_b2    = (const float*)d_in[15];
    const float* bn_g      = (const float*)d_in[16];
    const float* bn_b      = (const float*)d_in[17];
    const float* vn_W1     = (const float*)d_in[18];
    const float* vn_b1     = (const float*)d_in[19];
    const float* vn_bn1_g  = (const float*)d_in[20];
    const float* vn_bn1_b  = (const float*)d_in[21];
    const float* vn_W2     = (const float*)d_in[22];
    const float* vn_b2     = (const float*)d_in[23];
    const float* vn_bn2_g  = (const float*)d_in[24];
    const float* vn_bn2_b  = (const float*)d_in[25];
    float* out = (float*)d_out;

    // workspace carve-up
    float* ws     = (float*)d_ws;
    float* h      = ws;                      // N*D
    float* h_in   = h      + (size_t)Nn * Dn;
    float* Zb     = h_in   + (size_t)Nn * Dn;
    float* z1     = Zb     + (size_t)Nn * Dn;      // N*2D
    float* z2     = z1     + (size_t)Nn * D2n;     // N*D
    float* vn     = z2     + (size_t)Nn * Dn;      // G*D
    float* pooled = vn     + (size_t)Gn * Dn;      // G*D
    float* t1n    = pooled + (size_t)Gn * Dn;      // G*2D
    float* sums   = t1n    + (size_t)Gn * D2n;     // 2D
    float* sqs    = sums   + D2n;                  // 2D
    float* acoef  = sqs    + D2n;                  // 2D
    float* ccoef  = acoef  + D2n;                  // 2D

    const int totalND  = Nn * Dn;
    const int blocksND = (totalND + 255) / 256;

    // vn = vn_emb broadcast
    k_vn_init<<<(Gn * Dn + 255) / 256, 256, 0, stream>>>(vn_emb, vn, Gn * Dn);

    // encoder: h = x @ enc_W + enc_b   (M=20000, K=128, NC=128)
    {
        int tiles = (Nn / 16) * (Dn / 16);
        k_gemm_wmma<false><<<tiles / 8, 256, 0, stream>>>(x, enc_W, enc_b, h,
                                                          Nn, Dn, Dn, nullptr, nullptr);
    }

    for (int l = 0; l < Ln; ++l) {
        const float* eWl  = edge_W + (size_t)l * EDn * Dn;
        const float* ebl  = edge_b + (size_t)l * Dn;
        const float* W1l  = mlp_W1 + (size_t)l * Dn * D2n;
        const float* b1l  = mlp_b1 + (size_t)l * D2n;
        const float* g1l  = mlp_bn1_g + (size_t)l * D2n;
        const float* be1l = mlp_bn1_b + (size_t)l * D2n;
        const float* W2l  = mlp_W2 + (size_t)l * D2n * Dn;
        const float* b2l  = mlp_b2 + (size_t)l * Dn;
        const float* gol  = bn_g + (size_t)l * Dn;
        const float* bol  = bn_b + (size_t)l * Dn;

        // h_in = h + vn[batch]; Z = (1+eps_l)*h_in
        k_hin<<<blocksND, 256, 0, stream>>>(h, vn, batch, eps + l, h_in, Zb, totalND);

        // edge messages + scatter-add into Z
        k_edge<<<En / 8, 256, 0, stream>>>(edge_attr, eidx, eWl, ebl, h_in, Zb, En);

        // z1 = Z @ W1 + b1   (M=20000, K=128, NC=256)
        {
            int tiles = (Nn / 16) * (D2n / 16);
            k_gemm_wmma<false><<<tiles / 8, 256, 0, stream>>>(Zb, W1l, b1l, z1,
                                                              Nn, Dn, D2n, nullptr, nullptr);
        }

        // BN1 stats over N for 256 cols
        hipMemsetAsync(sums, 0, (size_t)2 * D2n * sizeof(float), stream);
        k_colstats<<<128, D2n, 0, stream>>>(z1, Nn, D2n, sums, sqs);
        k_bncoef<<<1, D2n, 0, stream>>>(sums, sqs, g1l, be1l, D2n, 1.0f / Nn, acoef, ccoef);

        // z2 = relu(bn(z1)) @ W2 + b2   (fused transform on A; M=20000, K=256, NC=128)
        {
            int tiles = (Nn / 16) * (Dn / 16);
            k_gemm_wmma<true><<<tiles / 8, 256, 0, stream>>>(z1, W2l, b2l, z2,
                                                             Nn, D2n, Dn, acoef, ccoef);
        }

        // outer BN stats over N for 128 cols
        hipMemsetAsync(sums, 0, (size_t)2 * D2n * sizeof(float), stream);
        k_colstats<<<128, Dn, 0, stream>>>(z2, Nn, Dn, sums, sqs);
        k_bncoef<<<1, Dn, 0, stream>>>(sums, sqs, gol, bol, Dn, 1.0f / Nn, acoef, ccoef);

        // hn = bn(z2), relu on all but last layer; last layer writes d_out
        float* dst = (l == Ln - 1) ? out : h;
        k_affine<<<blocksND, 256, 0, stream>>>(z2, acoef, ccoef, dst, totalND, Dn,
                                               (l < Ln - 1) ? 1 : 0);

        if (l < Ln - 1) {
            const float* vW1 = vn_W1 + (size_t)l * Dn * D2n;
            const float* vb1 = vn_b1 + (size_t)l * D2n;
            const float* vg1 = vn_bn1_g + (size_t)l * D2n;
            const float* vB1 = vn_bn1_b + (size_t)l * D2n;
            const float* vW2 = vn_W2 + (size_t)l * D2n * Dn;
            const float* vb2 = vn_b2 + (size_t)l * Dn;
            const float* vg2 = vn_bn2_g + (size_t)l * Dn;
            const float* vB2 = vn_bn2_b + (size_t)l * Dn;

            // pooled = vn + segment_sum(h_in, batch)
            k_pool_init<<<(Gn * Dn + 255) / 256, 256, 0, stream>>>(vn, pooled, Gn * Dn);
            k_pool_atomic<<<blocksND, 256, 0, stream>>>(h_in, batch, pooled, totalND);

            // t = relu(bn(pooled @ vn_W1 + b1)); vn = relu(bn(t @ vn_W2 + b2))
            k_vn_layer<<<D2n, Gn, 0, stream>>>(pooled, Dn, vW1, D2n, vb1, vg1, vB1, t1n);
            k_vn_layer<<<Dn, Gn, 0, stream>>>(t1n, D2n, vW2, Dn, vb2, vg2, vB2, vn);
        }
    }
}